// CombinedLoss_42322607735077
// MI455X (gfx1250) — compile-verified
//
#include <hip/hip_runtime.h>
#include <hip/hip_bf16.h>

typedef __attribute__((ext_vector_type(2))) float v2f;
typedef __attribute__((ext_vector_type(8))) float v8f;

#define B_   8
#define C_   8
#define H_   512
#define W_   1024
#define HW_  (H_ * W_)
#define NPIX_ (B_ * HW_)

// ws layout (floats): [0..63] sum_probs(b*8+c)  [64..127] inter  [128..191] onehot_cnt
//                     [192] ce_sum  [193] bd_sum   (194 floats total)

__device__ __forceinline__ float wave_sum32(float v) {
#pragma unroll
  for (int o = 16; o > 0; o >>= 1) v += __shfl_down(v, o, 32);
  return v;
}

__global__ void loss_init(float* __restrict__ ws) {
  int i = blockIdx.x * blockDim.x + threadIdx.x;
  if (i < 194) ws[i] = 0.f;
}

__global__ __launch_bounds__(256) void loss_main(const float* __restrict__ pred,
                                                 const int* __restrict__ tgt,
                                                 float* __restrict__ ws) {
  const int b = blockIdx.y;
  const float* predb = pred + (size_t)b * (size_t)(C_ * HW_);
  const int*   tgtb  = tgt  + (size_t)b * (size_t)HW_;

  float psum[C_], inter[C_], cnt[C_];
#pragma unroll
  for (int c = 0; c < C_; ++c) { psum[c] = 0.f; inter[c] = 0.f; cnt[c] = 0.f; }
  float ce_acc = 0.f, bd_acc = 0.f;

  const int stride = blockDim.x * gridDim.x;  // 32768 -> exactly 16 iters, no tail
  for (int pix = blockIdx.x * blockDim.x + threadIdx.x; pix < HW_; pix += stride) {
    // keep both streams flowing (global_prefetch_b8)
    __builtin_prefetch(tgtb  + pix + stride, 0, 0);
    __builtin_prefetch(predb + pix + stride, 0, 0);

    const int h = pix >> 10;
    const int w = pix & (W_ - 1);

    // 5x5 ellipse dilate/erode window; OOB taps skipped (padding never wins)
    const int t = tgtb[pix];
    int mn = t, mx = t;
    if (h >= 2)     { int v0 = tgtb[pix - 2 * W_]; mn = min(mn, v0); mx = max(mx, v0); }
    if (h + 2 < H_) { int v0 = tgtb[pix + 2 * W_]; mn = min(mn, v0); mx = max(mx, v0); }
#pragma unroll
    for (int dy = -1; dy <= 1; ++dy) {
      const int hh = h + dy;
      if (hh < 0 || hh >= H_) continue;
#pragma unroll
      for (int dx = -2; dx <= 2; ++dx) {
        const int ww = w + dx;
        if (ww < 0 || ww >= W_) continue;
        const int v0 = tgtb[hh * W_ + ww];
        mn = min(mn, v0); mx = max(mx, v0);
      }
    }
    const float wgt = (mx != mn) ? 10.f : 1.f;

    // fused softmax / CE / dice sums, one read of pred
    float v[C_];
#pragma unroll
    for (int c = 0; c < C_; ++c) v[c] = predb[(size_t)c * HW_ + pix];
    float m = v[0];
#pragma unroll
    for (int c = 1; c < C_; ++c) m = fmaxf(m, v[c]);
    float e[C_];
    float se = 0.f;
#pragma unroll
    for (int c = 0; c < C_; ++c) { e[c] = __expf(v[c] - m); se += e[c]; }
    const float inv = 1.f / se;
    const float lse = m + __logf(se);
    float vt = v[0];
#pragma unroll
    for (int c = 1; c < C_; ++c) vt = (t == c) ? v[c] : vt;
    const float ce = lse - vt;
    ce_acc += ce;
    bd_acc += ce * wgt;
#pragma unroll
    for (int c = 0; c < C_; ++c) {
      const float p = e[c] * inv;
      psum[c] += p;
      const bool is = (t == c);
      inter[c] += is ? p : 0.f;
      cnt[c]   += is ? 1.f : 0.f;
    }
  }

  // wave32 shuffle reduce -> LDS atomics -> one global atomic per bin per block
  __shared__ float s_acc[26];
  if (threadIdx.x < 26) s_acc[threadIdx.x] = 0.f;
  __syncthreads();

  const int lane = threadIdx.x & 31;
  float r;
#pragma unroll
  for (int c = 0; c < C_; ++c) {
    r = wave_sum32(psum[c]);  if (lane == 0) atomicAdd(&s_acc[c], r);
    r = wave_sum32(inter[c]); if (lane == 0) atomicAdd(&s_acc[8 + c], r);
    r = wave_sum32(cnt[c]);   if (lane == 0) atomicAdd(&s_acc[16 + c], r);
  }
  r = wave_sum32(ce_acc); if (lane == 0) atomicAdd(&s_acc[24], r);
  r = wave_sum32(bd_acc); if (lane == 0) atomicAdd(&s_acc[25], r);
  __syncthreads();

  const int tid = threadIdx.x;
  if (tid < 8)        atomicAdd(&ws[       b * 8 + tid],        s_acc[tid]);
  else if (tid < 16)  atomicAdd(&ws[64  +  b * 8 + (tid - 8)],  s_acc[tid]);
  else if (tid < 24)  atomicAdd(&ws[128 +  b * 8 + (tid - 16)], s_acc[tid]);
  else if (tid == 24) atomicAdd(&ws[192], s_acc[24]);
  else if (tid == 25) atomicAdd(&ws[193], s_acc[25]);
}

// Single wave32: dice ratios reduced via V_WMMA_F32_16X16X4_F32 (B = ones -> row sums).
// A 16x4 f32 layout: VGPR0 lanes0-15=K0 / lanes16-31=K2, VGPR1=K1/K3 (row M = lane%16).
__global__ void loss_final(const float* __restrict__ ws, float* __restrict__ out) {
  const int lane = threadIdx.x;
  const int row  = (lane < 16) ? lane : (lane - 16);
  const int kx   = (lane < 16) ? 0 : 2;
  const int bx   = row * 4 + kx;       // each of the 64 (b,c) bins covered once
  const int by   = bx + 1;
  const float sm = 1e-6f;
  const float rx = (2.f * ws[64 + bx] + sm) / (ws[bx] + ws[128 + bx] + sm);
  const float ry = (2.f * ws[64 + by] + sm) / (ws[by] + ws[128 + by] + sm);

  v2f a;    a.x = rx;  a.y = ry;
  v2f ones; ones.x = 1.f; ones.y = 1.f;
  v8f acc = {};
  acc = __builtin_amdgcn_wmma_f32_16x16x4_f32(false, a, false, ones,
                                              (short)0, acc, false, false);
  // D: lane j holds M=0..7 (VGPR0..7), lane j+16 holds M=8..15 -> finish with xor-16
  float s = 0.f;
#pragma unroll
  for (int i = 0; i < 8; ++i) s += acc[i];
  s += __shfl_xor(s, 16, 32);

  const float dice = 1.f - s * (1.f / 64.f);
  const float n    = (float)NPIX_;
  const float ce   = ws[192] / n;
  const float bd   = ws[193] / n;
  if (lane == 0) out[0] = 1.f * ce + 3.f * dice + 2.f * bd;
}

extern "C" void kernel_launch(void* const* d_in, const int* in_sizes, int n_in,
                              void* d_out, int out_size, void* d_ws, size_t ws_size,
                              hipStream_t stream) {
  const float* pred = (const float*)d_in[0];
  const int*   tgt  = (const int*)d_in[1];
  float* out = (float*)d_out;
  float* ws  = (float*)d_ws;

  loss_init<<<1, 256, 0, stream>>>(ws);
  loss_main<<<dim3(128, B_), 256, 0, stream>>>(pred, tgt, ws);
  loss_final<<<1, 32, 0, stream>>>(ws, out);
}